// SafeMoE_64673617543272
// MI455X (gfx1250) — compile-verified
//
#include <hip/hip_runtime.h>
#include <hip/hip_bf16.h>
#include <math.h>

// Problem constants (from reference)
#define T_TOK   8192
#define DM      1024      // D_MODEL
#define DF      4096      // D_FF
#define N_EXP   8
#define CAP     2560
// GEMM tiling
#define BM      64
#define BN      128
#define KT      32

typedef __attribute__((ext_vector_type(2))) float v2f;
typedef __attribute__((ext_vector_type(8))) float v8f;

__device__ __forceinline__ v8f wmma4(v2f a, v2f b, v8f c) {
    // D(16x16,f32) = A(16x4,f32) * B(4x16,f32) + C
    return __builtin_amdgcn_wmma_f32_16x16x4_f32(false, a, false, b, (short)0, c, false, false);
}

__device__ __forceinline__ float gelu_exact(float v) {
    return 0.5f * v * (1.0f + erff(v * 0.70710678118654752440f));
}

// ---------------------------------------------------------------- zero init
__global__ void zero_f32(float* __restrict__ p, int n) {
    int i = blockIdx.x * blockDim.x + threadIdx.x;
    if (i < n) p[i] = 0.0f;
}

// ---------------------------------------------------------------- router
// one wave (32 lanes) per token: logits = x[tok] @ Wr, softmax, top2, renorm
__global__ __launch_bounds__(256)
void moe_router(const float* __restrict__ x, const float* __restrict__ Wr,
                int2* __restrict__ eid, float2* __restrict__ esc) {
    const int lane = threadIdx.x & 31;
    const int wid  = threadIdx.x >> 5;
    const int tok  = blockIdx.x * 8 + wid;
    const float* xr = x + (size_t)tok * DM;

    float acc[N_EXP];
    #pragma unroll
    for (int e = 0; e < N_EXP; ++e) acc[e] = 0.0f;
    for (int d = lane; d < DM; d += 32) {
        float xv = xr[d];
        const float* wr = Wr + d * N_EXP;
        #pragma unroll
        for (int e = 0; e < N_EXP; ++e) acc[e] += xv * wr[e];
    }
    #pragma unroll
    for (int e = 0; e < N_EXP; ++e) {
        #pragma unroll
        for (int off = 16; off > 0; off >>= 1)
            acc[e] += __shfl_xor(acc[e], off);
    }
    if (lane == 0) {
        float mx = acc[0];
        #pragma unroll
        for (int e = 1; e < N_EXP; ++e) mx = fmaxf(mx, acc[e]);
        float p[N_EXP], s = 0.0f;
        #pragma unroll
        for (int e = 0; e < N_EXP; ++e) { p[e] = expf(acc[e] - mx); s += p[e]; }
        int i0 = 0;
        #pragma unroll
        for (int e = 1; e < N_EXP; ++e) if (p[e] > p[i0]) i0 = e;
        int i1 = (i0 == 0) ? 1 : 0;
        #pragma unroll
        for (int e = 0; e < N_EXP; ++e) if (e != i1 && e != i0 && p[e] > p[i1]) i1 = e;
        float p0 = p[i0] / s, p1 = p[i1] / s;
        float den = fmaxf(p0 + p1, 1e-9f);
        eid[tok] = make_int2(i0, i1);
        esc[tok] = make_float2(p0 / den, p1 / den);
    }
}

// ---------------------------------------------------------------- assignment
// Single workgroup reproduces the sequential per-expert capacity loop:
// rank = prefix-sum of eligibility in token order; keep if rank < CAP;
// overflow sets the fallback mask (affects later experts). Then compacts
// the fallback token list.
__global__ __launch_bounds__(1024)
void moe_assign(const int2* __restrict__ eid, const float2* __restrict__ esc,
                int* __restrict__ idxL, float* __restrict__ gateL,
                int* __restrict__ fbIdx, int* __restrict__ counts) {
    __shared__ unsigned char fb[T_TOK];
    __shared__ int waveTot[32];
    __shared__ int waveOff[32];
    __shared__ int chunkTot;
    __shared__ int run;

    const int t    = threadIdx.x;
    const int lane = t & 31;
    const int wid  = t >> 5;

    for (int i = t; i < T_TOK; i += 1024) fb[i] = 0;
    __syncthreads();

    for (int e = 0; e < N_EXP; ++e) {
        if (t == 0) run = 0;
        __syncthreads();
        for (int base = 0; base < T_TOK; base += 1024) {
            const int tok = base + t;
            int2   ex = eid[tok];
            float2 sc = esc[tok];
            bool a0 = (ex.x == e);
            bool a1 = (ex.y == e);
            bool elig = (a0 || a1) && (fb[tok] == 0);
            unsigned mask = (unsigned)__ballot(elig);
            int rw = __popc(mask & ((1u << lane) - 1u));
            if (lane == 0) waveTot[wid] = __popc(mask);
            __syncthreads();
            if (wid == 0) {
                int v = waveTot[lane];
                int xi = v;
                for (int d = 1; d < 32; d <<= 1) {
                    int o = __shfl_up(xi, d);
                    if (lane >= d) xi += o;
                }
                waveOff[lane] = xi - v;
                if (lane == 31) chunkTot = xi;
            }
            __syncthreads();
            int rank = run + waveOff[wid] + rw;
            if (elig) {
                if (rank < CAP) {
                    idxL[e * CAP + rank]  = tok;
                    gateL[e * CAP + rank] = a0 ? sc.x : sc.y;
                } else {
                    fb[tok] = 1;   // overflow -> fallback for experts > e
                }
            }
            __syncthreads();
            if (t == 0) run += chunkTot;
            __syncthreads();
        }
        if (t == 0) counts[e] = (run < CAP) ? run : CAP;
        __syncthreads();
    }

    // compact fallback token list
    if (t == 0) run = 0;
    __syncthreads();
    for (int base = 0; base < T_TOK; base += 1024) {
        const int tok = base + t;
        bool elig = (fb[tok] != 0);
        unsigned mask = (unsigned)__ballot(elig);
        int rw = __popc(mask & ((1u << lane) - 1u));
        if (lane == 0) waveTot[wid] = __popc(mask);
        __syncthreads();
        if (wid == 0) {
            int v = waveTot[lane];
            int xi = v;
            for (int d = 1; d < 32; d <<= 1) {
                int o = __shfl_up(xi, d);
                if (lane >= d) xi += o;
            }
            waveOff[lane] = xi - v;
            if (lane == 31) chunkTot = xi;
        }
        __syncthreads();
        int rank = run + waveOff[wid] + rw;
        if (elig) fbIdx[rank] = tok;
        __syncthreads();
        if (t == 0) run += chunkTot;
        __syncthreads();
    }
    if (t == 0) counts[8] = run;
}

// ---------------------------------------------------------------- GEMM1
// h[m, 0:DF] = gelu( x[idx[m], :] @ W[DM, DF] + bias )   for m < cnt
__global__ __launch_bounds__(256)
void ffn_gemm1(const float* __restrict__ xg, const float* __restrict__ W,
               const float* __restrict__ bias, const int* __restrict__ idxList,
               const int* __restrict__ cntPtr, int cntSub,
               float* __restrict__ h) {
    __shared__ float Xs[BM][KT + 2];
    __shared__ float Ws[BN][KT + 2];
    __shared__ int   tokS[BM];

    int cnt = *cntPtr - cntSub;
    cnt = cnt < 0 ? 0 : (cnt > CAP ? CAP : cnt);
    const int m0 = blockIdx.y * BM;
    if (m0 >= cnt) return;
    const int n0   = blockIdx.x * BN;
    const int t    = threadIdx.x;
    const int lane = t & 31;
    const int w    = t >> 5;
    const int wm   = w & 1;     // 2 waves over M
    const int wn   = w >> 1;    // 4 waves over N
    const int mrow = lane & 15;
    const int kb   = (lane >> 4) << 1;

    if (t < BM) tokS[t] = (m0 + t < cnt) ? idxList[m0 + t] : 0;
    __syncthreads();

    v8f acc00 = {}, acc01 = {}, acc10 = {}, acc11 = {};

    for (int k0 = 0; k0 < DM; k0 += KT) {
        #pragma unroll
        for (int i = t; i < BM * KT; i += 256) {
            int m = i >> 5, kk = i & (KT - 1);
            Xs[m][kk] = xg[tokS[m] * DM + k0 + kk];
        }
        #pragma unroll
        for (int i = t; i < KT * BN; i += 256) {
            int kk = i >> 7, n = i & (BN - 1);
            Ws[n][kk] = W[(k0 + kk) * DF + n0 + n];
        }
        __syncthreads();
        #pragma unroll
        for (int kk = 0; kk < KT; kk += 4) {
            v2f a0  = *(const v2f*)&Xs[wm * 32 + mrow][kk + kb];
            v2f a1  = *(const v2f*)&Xs[wm * 32 + 16 + mrow][kk + kb];
            v2f b0  = *(const v2f*)&Ws[wn * 32 + mrow][kk + kb];
            v2f b1v = *(const v2f*)&Ws[wn * 32 + 16 + mrow][kk + kb];
            acc00 = wmma4(a0, b0, acc00);
            acc01 = wmma4(a0, b1v, acc01);
            acc10 = wmma4(a1, b0, acc10);
            acc11 = wmma4(a1, b1v, acc11);
        }
        __syncthreads();
    }

    const int rr = (lane >> 4) * 8;
    #pragma unroll
    for (int i = 0; i < 2; ++i) {
        #pragma unroll
        for (int j = 0; j < 2; ++j) {
            v8f a = (i == 0) ? (j == 0 ? acc00 : acc01) : (j == 0 ? acc10 : acc11);
            int cg = n0 + wn * 32 + j * 16 + mrow;
            float bb = bias[cg];
            #pragma unroll
            for (int v = 0; v < 8; ++v) {
                int m = m0 + wm * 32 + i * 16 + rr + v;
                if (m < cnt) h[m * DF + cg] = gelu_exact(a[v] + bb);
            }
        }
    }
}

// ---------------------------------------------------------------- GEMM2
// y[idx[m], :] (+)= gate[m] * ( h[m, :] @ W[DF, DM] + bias )   for m < cnt
__global__ __launch_bounds__(256)
void ffn_gemm2(const float* __restrict__ h, const float* __restrict__ W,
               const float* __restrict__ bias, const int* __restrict__ idxList,
               const float* __restrict__ gateList, const int* __restrict__ cntPtr,
               int cntSub, int addMode, float* __restrict__ y) {
    __shared__ float Xs[BM][KT + 2];
    __shared__ float Ws[BN][KT + 2];
    __shared__ int   tokS[BM];
    __shared__ float gateS[BM];

    int cnt = *cntPtr - cntSub;
    cnt = cnt < 0 ? 0 : (cnt > CAP ? CAP : cnt);
    const int m0 = blockIdx.y * BM;
    if (m0 >= cnt) return;
    const int n0   = blockIdx.x * BN;
    const int t    = threadIdx.x;
    const int lane = t & 31;
    const int w    = t >> 5;
    const int wm   = w & 1;
    const int wn   = w >> 1;
    const int mrow = lane & 15;
    const int kb   = (lane >> 4) << 1;

    if (t < BM) {
        bool ok = (m0 + t < cnt);
        tokS[t]  = ok ? idxList[m0 + t] : 0;
        gateS[t] = gateList ? (ok ? gateList[m0 + t] : 0.0f) : 1.0f;
    }
    __syncthreads();

    v8f acc00 = {}, acc01 = {}, acc10 = {}, acc11 = {};

    for (int k0 = 0; k0 < DF; k0 += KT) {
        #pragma unroll
        for (int i = t; i < BM * KT; i += 256) {
            int m = i >> 5, kk = i & (KT - 1);
            Xs[m][kk] = h[(m0 + m) * DF + k0 + kk];
        }
        #pragma unroll
        for (int i = t; i < KT * BN; i += 256) {
            int kk = i >> 7, n = i & (BN - 1);
            Ws[n][kk] = W[(k0 + kk) * DM + n0 + n];
        }
        __syncthreads();
        #pragma unroll
        for (int kk = 0; kk < KT; kk += 4) {
            v2f a0  = *(const v2f*)&Xs[wm * 32 + mrow][kk + kb];
            v2f a1  = *(const v2f*)&Xs[wm * 32 + 16 + mrow][kk + kb];
            v2f b0  = *(const v2f*)&Ws[wn * 32 + mrow][kk + kb];
            v2f b1v = *(const v2f*)&Ws[wn * 32 + 16 + mrow][kk + kb];
            acc00 = wmma4(a0, b0, acc00);
            acc01 = wmma4(a0, b1v, acc01);
            acc10 = wmma4(a1, b0, acc10);
            acc11 = wmma4(a1, b1v, acc11);
        }
        __syncthreads();
    }

    const int rr = (lane >> 4) * 8;
    #pragma unroll
    for (int i = 0; i < 2; ++i) {
        #pragma unroll
        for (int j = 0; j < 2; ++j) {
            v8f a = (i == 0) ? (j == 0 ? acc00 : acc01) : (j == 0 ? acc10 : acc11);
            int cg = n0 + wn * 32 + j * 16 + mrow;
            float bb = bias[cg];
            #pragma unroll
            for (int v = 0; v < 8; ++v) {
                int lm = wm * 32 + i * 16 + rr + v;
                int m  = m0 + lm;
                if (m < cnt) {
                    float val = gateS[lm] * (a[v] + bb);
                    float* dst = &y[(size_t)tokS[lm] * DM + cg];
                    if (addMode) *dst += val; else *dst = val;
                }
            }
        }
    }
}

// ---------------------------------------------------------------- launch
extern "C" void kernel_launch(void* const* d_in, const int* in_sizes, int n_in,
                              void* d_out, int out_size, void* d_ws, size_t ws_size,
                              hipStream_t stream) {
    const float* x   = (const float*)d_in[0];
    const float* Wr  = (const float*)d_in[1];
    const float* W1  = (const float*)d_in[2];
    const float* b1  = (const float*)d_in[3];
    const float* W2  = (const float*)d_in[4];
    const float* b2  = (const float*)d_in[5];
    const float* fw1 = (const float*)d_in[6];
    const float* fb1 = (const float*)d_in[7];
    const float* fw2 = (const float*)d_in[8];
    const float* fb2 = (const float*)d_in[9];
    float* y = (float*)d_out;
    (void)in_sizes; (void)n_in; (void)out_size; (void)ws_size;

    char* ws = (char*)d_ws;
    size_t off = 0;
    auto alloc = [&](size_t bytes) -> void* {
        void* p = ws + off;
        off = (off + bytes + 255) & ~(size_t)255;
        return p;
    };
    float*  hbuf   = (float*) alloc((size_t)CAP * DF * sizeof(float));   // 41.9 MB
    int2*   eid    = (int2*)  alloc((size_t)T_TOK * sizeof(int2));
    float2* esc    = (float2*)alloc((size_t)T_TOK * sizeof(float2));
    int*    idxL   = (int*)   alloc((size_t)N_EXP * CAP * sizeof(int));
    float*  gateL  = (float*) alloc((size_t)N_EXP * CAP * sizeof(float));
    int*    fbIdx  = (int*)   alloc((size_t)T_TOK * sizeof(int));
    int*    counts = (int*)   alloc(16 * sizeof(int));

    // 1) zero output (scatter-add target)
    zero_f32<<<(T_TOK * DM) / 256, 256, 0, stream>>>(y, T_TOK * DM);
    // 2) router
    moe_router<<<T_TOK / 8, 256, 0, stream>>>(x, Wr, eid, esc);
    // 3) sequential capacity assignment + fallback compaction
    moe_assign<<<1, 1024, 0, stream>>>(eid, esc, idxL, gateL, fbIdx, counts);

    // 4) experts: gather -> gelu(x@W1+b1) -> @W2+b2, gate-scaled scatter-add
    dim3 g1(DF / BN, CAP / BM);   // 32 x 40
    dim3 g2(DM / BN, CAP / BM);   //  8 x 40
    for (int e = 0; e < N_EXP; ++e) {
        ffn_gemm1<<<g1, 256, 0, stream>>>(x, W1 + (size_t)e * DM * DF,
                                          b1 + (size_t)e * DF,
                                          idxL + e * CAP, counts + e, 0, hbuf);
        ffn_gemm2<<<g2, 256, 0, stream>>>(hbuf, W2 + (size_t)e * DF * DM,
                                          b2 + (size_t)e * DM,
                                          idxL + e * CAP, gateL + e * CAP,
                                          counts + e, 0, /*add=*/1, y);
    }
    // 5) fallback FFN last, overwriting fb tokens (matches reference overwrite
    //    semantics); process fb list in 4 chunks of CAP rows through hbuf
    for (int c = 0; c < 4; ++c) {
        ffn_gemm1<<<g1, 256, 0, stream>>>(x, fw1, fb1, fbIdx + c * CAP,
                                          counts + 8, c * CAP, hbuf);
        ffn_gemm2<<<g2, 256, 0, stream>>>(hbuf, fw2, fb2, fbIdx + c * CAP,
                                          /*gate=*/nullptr, counts + 8, c * CAP,
                                          /*add=*/0, y);
    }
}